// HRM_47184510714066
// MI455X (gfx1250) — compile-verified
//
#include <hip/hip_runtime.h>
#include <hip/hip_bf16.h>
#include <float.h>

typedef __attribute__((ext_vector_type(2))) float v2f;
typedef __attribute__((ext_vector_type(8))) float v8f;

#define WAVES_PER_BLOCK 16
#define HSTRIDE 68   // padded LDS row stride (floats): bank = (4*row+col)%64 -> conflict-free
#define MAXT 96      // supports LL+LS <= 96 (actual T = 70)
#define PFD 8        // gather prefetch distance (items)

__global__ __launch_bounds__(32 * WAVES_PER_BLOCK)
void hrm_forward_kernel(const int* __restrict__ user_inputs,
                        const int* __restrict__ L,
                        const int* __restrict__ S,
                        const int* __restrict__ item_inputs,
                        const float* __restrict__ user_table,
                        const float* __restrict__ item_table,
                        float* __restrict__ out,
                        int B, int LL, int LS)
{
  __shared__ int   cids[WAVES_PER_BLOCK][MAXT];     // compacted valid ids per sample
  __shared__ float hyb [WAVES_PER_BLOCK][HSTRIDE];  // hybrid vectors   [16][64] (padded)
  __shared__ float qe  [WAVES_PER_BLOCK][HSTRIDE];  // query item embed [16][64] (padded)

  const int  lane = threadIdx.x & 31;
  const int  w    = __builtin_amdgcn_readfirstlane(threadIdx.x >> 5); // uniform wave id
  const int  s    = __builtin_amdgcn_readfirstlane(
                        blockIdx.x * WAVES_PER_BLOCK + w);            // uniform sample id
  const bool active = (s < B);
  const int  T      = LL + LS;

  // ---- Phase A: stable compaction of valid item ids into LDS (one wave/sample)
  const int* Lr = L + (long long)s * LL;
  const int* Sr = S + (long long)s * LS;
  const unsigned lt = (1u << lane) - 1u;
  int n = 0;
  #pragma unroll
  for (int seg = 0; seg < (MAXT / 32); ++seg) {
    int p  = lane + seg * 32;
    int id = -1;
    if (active && p < T) id = (p < LL) ? Lr[p] : Sr[p - LL];
    unsigned bb = __builtin_amdgcn_ballot_w32(id >= 0);      // wave32 ballot
    if (id >= 0) cids[w][n + __popc(bb & lt)] = id;          // stable rank
    n += __popc(bb);
  }
  // iterated MaxPool1d(2) reduces over only the first m = 2^floor(log2(n)) items
  const int m = (n > 0) ? (1 << (31 - __clz(n))) : 0;

  // ---- Phase B: max-pool first m gathered item rows.
  // lane owns dims [2*lane, 2*lane+1]: one wave load == one coalesced 256B row.
  float mx0 = -FLT_MAX, mx1 = -FLT_MAX;
  int t = 0;
  #pragma unroll 4
  for (; t < m - PFD; ++t) {                                 // main: prefetch unconditionally
    __builtin_prefetch(item_table + (long long)cids[w][t + PFD] * 64 + 2 * lane, 0, 1);
    long long id = cids[w][t];                               // uniform LDS broadcast
    float2 e = *(const float2*)(item_table + id * 64 + 2 * lane);
    mx0 = fmaxf(mx0, e.x);
    mx1 = fmaxf(mx1, e.y);
  }
  #pragma unroll 4
  for (; t < m; ++t) {                                       // tail: no prefetch
    long long id = cids[w][t];
    float2 e = *(const float2*)(item_table + id * 64 + 2 * lane);
    mx0 = fmaxf(mx0, e.x);
    mx1 = fmaxf(mx1, e.y);
  }

  if (active) {
    long long uid = user_inputs[s];                          // uniform -> s_load
    long long qid = item_inputs[s];                          // uniform -> s_load
    float2 u = *(const float2*)(user_table + uid * 64 + 2 * lane);
    float2 q = *(const float2*)(item_table + qid * 64 + 2 * lane);
    hyb[w][2 * lane + 0] = fmaxf(mx0, u.x);   // hybrid = max(user, item_pooling)
    hyb[w][2 * lane + 1] = fmaxf(mx1, u.y);
    qe [w][2 * lane + 0] = q.x;
    qe [w][2 * lane + 1] = q.y;
  } else {
    hyb[w][2 * lane + 0] = 0.f; hyb[w][2 * lane + 1] = 0.f;
    qe [w][2 * lane + 0] = 0.f; qe [w][2 * lane + 1] = 0.f;
  }
  __syncthreads();

  // ---- Phase C (wave 0 only, EXEC all-ones): scores = diag(QE x HYB^T)
  // 16 x V_WMMA_F32_16X16X4_F32 accumulating over K = 64.
  if (w == 0) {
    v8f c = {};
    const int row  = lane & 15;            // A: M index / B: N index
    const int koff = (lane >> 4) << 1;     // lanes 0-15 -> K{0,1}; lanes 16-31 -> K{2,3}
    #pragma unroll
    for (int kc = 0; kc < 16; ++kc) {
      v2f a = *(const v2f*)&qe [row][kc * 4 + koff];  // A 16x4 chunk (per-ISA layout)
      v2f b = *(const v2f*)&hyb[row][kc * 4 + koff];  // B 4x16 chunk = HYB^T
      c = __builtin_amdgcn_wmma_f32_16x16x4_f32(false, a, false, b,
                                                (short)0, c, false, false);
    }
    // Branchless diagonal extraction.
    // D[i][i]: lane i holds c[i] (i<8); lane i+16 holds c[i-8] (i>=8).
    const int k = lane & 7;
    float d = c[0];
    #pragma unroll
    for (int v = 1; v < 8; ++v) d = (k == v) ? c[v] : d;     // v_cndmask chain
    const bool lo = (lane < 8), hi = (lane >= 24);
    const long long base = (long long)blockIdx.x * WAVES_PER_BLOCK;
    const long long oidx = base + (lo ? lane : lane - 16);
    if ((lo || hi) && oidx < B) out[oidx] = d;               // single coalesced store
  }
}

extern "C" void kernel_launch(void* const* d_in, const int* in_sizes, int n_in,
                              void* d_out, int out_size, void* d_ws, size_t ws_size,
                              hipStream_t stream) {
  const int*   user_inputs = (const int*)  d_in[0];
  const int*   L           = (const int*)  d_in[1];
  const int*   S           = (const int*)  d_in[2];
  const int*   item_inputs = (const int*)  d_in[3];
  const float* user_table  = (const float*)d_in[4];
  const float* item_table  = (const float*)d_in[5];
  float*       out         = (float*)d_out;

  const int B  = in_sizes[0];
  const int LL = in_sizes[1] / B;
  const int LS = in_sizes[2] / B;

  const int blocks = (B + WAVES_PER_BLOCK - 1) / WAVES_PER_BLOCK;
  hipLaunchKernelGGL(hrm_forward_kernel, dim3(blocks), dim3(32 * WAVES_PER_BLOCK),
                     0, stream,
                     user_inputs, L, S, item_inputs, user_table, item_table,
                     out, B, LL, LS);
}